// TripleViewNet_43009802502724
// MI455X (gfx1250) — compile-verified
//
#include <hip/hip_runtime.h>
#include <math.h>

// ---------------- problem constants (fixed by reference) ----------------
#define NN 100000
#define EE 1600000
#define FF 256
#define GG 512
#define HC 64          // H*C = 2*32
#define HALL 192       // 3 views * 64
#define LDS_STRIDE 260 // 256 + 4 pad to avoid bank conflicts
#define ITEMS (EE + NN) // edges + self loops

typedef float v2f __attribute__((ext_vector_type(2)));
typedef float v8f __attribute__((ext_vector_type(8)));

// ---------------- atomic helpers ----------------
__device__ __forceinline__ void atomAddF(float* p, float v) {
    __hip_atomic_fetch_add(p, v, __ATOMIC_RELAXED, __HIP_MEMORY_SCOPE_AGENT);
}
__device__ __forceinline__ void atomMaxF(float* p, float v) {
    if (v >= 0.0f) atomicMax((int*)p, __float_as_int(v));
    else           atomicMin((unsigned int*)p, __float_as_uint(v));
}

// ---------------- fill ----------------
__global__ void fill_kernel(float* __restrict__ p, long long n, float v) {
    long long i = (long long)blockIdx.x * blockDim.x + threadIdx.x;
    long long stride = (long long)gridDim.x * blockDim.x;
    for (; i < n; i += stride) p[i] = v;
}

// ---------------- BatchNorm stats: per-column sum / sumsq ----------------
__global__ void bn_stats_kernel(const float* __restrict__ x,
                                float* __restrict__ colsum, float* __restrict__ colsq) {
    int col = threadIdx.x;                 // 256 threads = 256 columns
    int r0 = blockIdx.x * 128;
    int rend = r0 + 128; if (rend > NN) rend = NN;
    float s = 0.f, q = 0.f;
    for (int r = r0; r < rend; ++r) {
        float v = x[(size_t)r * FF + col];
        s += v; q += v * v;
    }
    atomAddF(&colsum[col], s);
    atomAddF(&colsq[col], q);
}

__global__ void bn_finalize_kernel(const float* __restrict__ colsum,
                                   const float* __restrict__ colsq,
                                   const float* __restrict__ bn_w,
                                   const float* __restrict__ bn_b,
                                   float* __restrict__ scale, float* __restrict__ shift) {
    int c = threadIdx.x;
    float mean = colsum[c] * (1.0f / NN);
    float var  = colsq[c] * (1.0f / NN) - mean * mean;
    float sc   = bn_w[c] / sqrtf(var + 1e-5f);
    scale[c] = sc;
    shift[c] = bn_b[c] - mean * sc;
}

// ---------------- W transpose: [256,64] -> [64,256] (k contiguous) ----------
__global__ void wtrans_kernel(const float* __restrict__ W, float* __restrict__ Wt) {
    int idx = blockIdx.x * blockDim.x + threadIdx.x;
    if (idx >= FF * HC) return;
    int k = idx >> 6, c = idx & 63;
    Wt[c * FF + k] = W[idx];
}

// ---------------- fused BN-apply + triple GEMM via WMMA fp32 ----------------
// Block = 128 threads (4 waves). Block handles 16 rows; wave w handles local
// col-tile w (cols 16w..16w+15) of each of the 3 view weight matrices.
__global__ void __launch_bounds__(128)
gemm_bn_wmma_kernel(const float* __restrict__ x,
                    const float* __restrict__ scale, const float* __restrict__ shift,
                    const float* __restrict__ Wta, const float* __restrict__ Wtc,
                    const float* __restrict__ Wtp, float* __restrict__ h_all) {
    __shared__ float lds[16 * LDS_STRIDE];
    const int row0 = blockIdx.x * 16;
    const int tid = threadIdx.x;

    // stage BN'd 16x256 tile of x into LDS (coalesced)
    for (int idx = tid; idx < 16 * 256; idx += 128) {
        int r = idx >> 8, c = idx & 255;
        float v = x[(size_t)(row0 + r) * FF + c];
        lds[r * LDS_STRIDE + c] = v * scale[c] + shift[c];
    }
    __syncthreads();

    const int wave  = tid >> 5;         // 0..3 -> col tile within each view
    const int lane  = tid & 31;
    const int m16   = lane & 15;        // A row / B col within tile
    const int khalf = (lane >> 4) << 1; // lanes>=16 handle K+2,K+3
    const int cloc  = wave * 16 + m16;  // local col 0..63 inside a view's W

    v8f acc0 = {}, acc1 = {}, acc2 = {};
    const float* lrow = &lds[m16 * LDS_STRIDE];
    const float* brow0 = Wta + (size_t)cloc * FF;
    const float* brow1 = Wtc + (size_t)cloc * FF;
    const float* brow2 = Wtp + (size_t)cloc * FF;

    for (int k = 0; k < 256; k += 4) {
        const int kb = k + khalf;
        v2f a  = *(const v2f*)(lrow + kb);   // A[m16][kb], A[m16][kb+1]
        v2f b0 = *(const v2f*)(brow0 + kb);  // W[kb][cloc], W[kb+1][cloc]
        v2f b1 = *(const v2f*)(brow1 + kb);
        v2f b2 = *(const v2f*)(brow2 + kb);
        acc0 = __builtin_amdgcn_wmma_f32_16x16x4_f32(false, a, false, b0, (short)0, acc0, false, false);
        acc1 = __builtin_amdgcn_wmma_f32_16x16x4_f32(false, a, false, b1, (short)0, acc1, false, false);
        acc2 = __builtin_amdgcn_wmma_f32_16x16x4_f32(false, a, false, b2, (short)0, acc2, false, false);
    }

    // D layout: VGPR g, lanes 0-15 -> M=g ; lanes 16-31 -> M=8+g ; N = lane&15
    const int mbase = (lane >> 4) * 8;
    const int ncol  = wave * 16 + m16;
    for (int g = 0; g < 8; ++g) {
        size_t rbase = (size_t)(row0 + mbase + g) * HALL;
        h_all[rbase +   0 + ncol] = acc0[g];
        h_all[rbase +  64 + ncol] = acc1[g];
        h_all[rbase + 128 + ncol] = acc2[g];
    }
}

// ---------------- attention projections: a_src/a_dst [3][N][2] ----------------
__global__ void attn_kernel(const float* __restrict__ h_all,
                            const float* __restrict__ as0, const float* __restrict__ ad0,
                            const float* __restrict__ as1, const float* __restrict__ ad1,
                            const float* __restrict__ as2, const float* __restrict__ ad2,
                            float* __restrict__ a_src, float* __restrict__ a_dst) {
    int t = blockIdx.x * blockDim.x + threadIdx.x;
    if (t >= 3 * NN * 2) return;
    int v = t / (NN * 2);
    int r = t - v * (NN * 2);
    int n = r >> 1, h = r & 1;
    const float* as = (v == 0 ? as0 : v == 1 ? as1 : as2) + h * 32;
    const float* ad = (v == 0 ? ad0 : v == 1 ? ad1 : ad2) + h * 32;
    const float* hv = h_all + (size_t)n * HALL + v * 64 + h * 32;
    float ss = 0.f, sd = 0.f;
    #pragma unroll
    for (int c = 0; c < 32; ++c) { float hh = hv[c]; ss += hh * as[c]; sd += hh * ad[c]; }
    a_src[t] = ss;
    a_dst[t] = sd;
}

// ---------------- CSR build pass 1: in-degree histogram ----------------
__global__ void count_kernel(const int* __restrict__ ei, int* __restrict__ count) {
    int idx = blockIdx.x * blockDim.x + threadIdx.x;
    if (idx >= ITEMS) return;
    int dst = (idx < EE) ? ei[EE + idx] : (idx - EE);
    atomicAdd(&count[dst], 1);
}

// ---------------- CSR build pass 2: exclusive scan (single block) ----------
__global__ void __launch_bounds__(1024)
scan_kernel(const int* __restrict__ count, int* __restrict__ offs, int* __restrict__ cursor) {
    __shared__ int sums[1024];
    const int t = threadIdx.x;
    const int chunk = (NN + 1023) / 1024;
    int beg = t * chunk, end = beg + chunk; if (end > NN) end = NN; if (beg > NN) beg = NN;
    int s = 0;
    for (int i = beg; i < end; ++i) s += count[i];
    sums[t] = s;
    __syncthreads();
    for (int d = 1; d < 1024; d <<= 1) {
        int u = (t >= d) ? sums[t - d] : 0;
        __syncthreads();
        sums[t] += u;
        __syncthreads();
    }
    int run = (t == 0) ? 0 : sums[t - 1];
    for (int i = beg; i < end; ++i) {
        offs[i] = run; cursor[i] = run;
        run += count[i];
    }
    if (t == 1023) offs[NN] = sums[1023];   // total = ITEMS
}

// ---------------- CSR build pass 3: scatter src ids ----------------
__global__ void scatter_kernel(const int* __restrict__ ei, int* __restrict__ cursor,
                               int* __restrict__ csr_src) {
    int idx = blockIdx.x * blockDim.x + threadIdx.x;
    if (idx >= ITEMS) return;
    int src, dst;
    if (idx < EE) { src = ei[idx]; dst = ei[EE + idx]; }
    else          { src = dst = idx - EE; }
    int pos = atomicAdd(&cursor[dst], 1);
    csr_src[pos] = src;
}

// ---------------- dst-centric softmax-attention gather (wave per node) -------
// Fuses: segment max, exp-sum, weighted aggregation, +bias, ELU, and
// segment-max pooling into gfeat (graph features).
__global__ void gather_kernel(const int* __restrict__ offs, const int* __restrict__ csr_src,
                              const float* __restrict__ aS, const float* __restrict__ aD,
                              const float* __restrict__ h_all, int voff,
                              const float* __restrict__ bias, const int* __restrict__ batch,
                              float* __restrict__ gfeat) {
    int gid = blockIdx.x * blockDim.x + threadIdx.x;
    int dst = gid >> 5;
    int lane = gid & 31;
    if (dst >= NN) return;
    const int beg = offs[dst], end = offs[dst + 1];
    const int h = lane >> 4;               // features 2*lane,2*lane+1 -> head lane/16
    const float adst = aD[dst * 2 + h];

    // pass 1: segment max of leaky-relu scores
    float mx = -INFINITY;
    for (int i = beg; i < end; ++i) {
        float e = aS[csr_src[i] * 2 + h] + adst;
        e = e > 0.f ? e : 0.2f * e;
        mx = fmaxf(mx, e);
    }
    // pass 2: denominator
    float den = 0.f;
    for (int i = beg; i < end; ++i) {
        float e = aS[csr_src[i] * 2 + h] + adst;
        e = e > 0.f ? e : 0.2f * e;
        den += expf(e - mx);
    }
    const float dinv = 1.0f / (den + 1e-16f);
    // pass 3: weighted feature aggregation (2 features per lane)
    float acc0 = 0.f, acc1 = 0.f;
    for (int i = beg; i < end; ++i) {
        int s = csr_src[i];
        float e = aS[s * 2 + h] + adst;
        e = e > 0.f ? e : 0.2f * e;
        float alpha = expf(e - mx) * dinv;
        const float* hs = h_all + (size_t)s * HALL + voff + lane * 2;
        acc0 += hs[0] * alpha;
        acc1 += hs[1] * alpha;
    }
    // epilogue: +bias, ELU, pool (segment max over batch)
    int f0 = lane * 2;
    float v0 = acc0 + bias[f0];
    float v1 = acc1 + bias[f0 + 1];
    v0 = v0 > 0.f ? v0 : (expf(v0) - 1.0f);
    v1 = v1 > 0.f ? v1 : (expf(v1) - 1.0f);
    float* gf = gfeat + (size_t)batch[dst] * HC + f0;
    atomMaxF(gf,     v0);
    atomMaxF(gf + 1, v1);
}

// ---------------- per-graph gating + classifier head ----------------
__global__ void head_kernel(const float* __restrict__ gfeat,
                            const float* __restrict__ g1w, const float* __restrict__ g1b,
                            const float* __restrict__ g2w, const float* __restrict__ g2b,
                            const float* __restrict__ c1w, const float* __restrict__ c1b,
                            const float* __restrict__ c2w, const float* __restrict__ c2b,
                            const float* __restrict__ c3w, const float* __restrict__ c3b,
                            float* __restrict__ out) {
    int g = blockIdx.x * blockDim.x + threadIdx.x;
    if (g >= GG) return;
    const float* ha = gfeat + (size_t)g * HC;
    const float* hc = gfeat + (size_t)GG * HC + (size_t)g * HC;
    const float* hp = gfeat + (size_t)2 * GG * HC + (size_t)g * HC;

    float avg[64];
    for (int f = 0; f < 64; ++f) avg[f] = (ha[f] + hc[f] + hp[f]) / 3.0f;

    float t[32];
    for (int j = 0; j < 32; ++j) {
        float s = g1b[j];
        for (int k = 0; k < 64; ++k) s += avg[k] * g1w[k * 32 + j];
        t[j] = s > 0.f ? s : 0.f;
    }
    float l3[3];
    for (int o = 0; o < 3; ++o) {
        float s = g2b[o];
        for (int j = 0; j < 32; ++j) s += t[j] * g2w[j * 3 + o];
        l3[o] = s;
    }
    float mx = fmaxf(l3[0], fmaxf(l3[1], l3[2]));
    float e0 = expf(l3[0] - mx), e1 = expf(l3[1] - mx), e2 = expf(l3[2] - mx);
    float inv = 1.0f / (e0 + e1 + e2);
    float w0 = e0 * inv, w1 = e1 * inv, w2 = e2 * inv;

    float fused[64];
    for (int f = 0; f < 64; ++f) fused[f] = w0 * ha[f] + w1 * hc[f] + w2 * hp[f];

    float z1[128];
    for (int j = 0; j < 128; ++j) {
        float s = c1b[j];
        for (int k = 0; k < 64; ++k) s += fused[k] * c1w[k * 128 + j];
        z1[j] = s > 0.f ? s : 0.f;
    }
    float z2[64];
    for (int j = 0; j < 64; ++j) {
        float s = c2b[j];
        for (int k = 0; k < 128; ++k) s += z1[k] * c2w[k * 64 + j];
        z2[j] = s > 0.f ? s : 0.f;
    }
    float s = c3b[0];
    for (int k = 0; k < 64; ++k) s += z2[k] * c3w[k];
    out[g] = s;
}

// ---------------- host launcher ----------------
extern "C" void kernel_launch(void* const* d_in, const int* in_sizes, int n_in,
                              void* d_out, int out_size, void* d_ws, size_t ws_size,
                              hipStream_t stream) {
    const float* x     = (const float*)d_in[0];
    const int*   ei[3] = { (const int*)d_in[1], (const int*)d_in[2], (const int*)d_in[3] };
    const int*   batch = (const int*)d_in[4];
    const float* bn_w  = (const float*)d_in[6];
    const float* bn_b  = (const float*)d_in[7];
    const float* W[3]     = { (const float*)d_in[8],  (const float*)d_in[12], (const float*)d_in[16] };
    const float* attS[3]  = { (const float*)d_in[9],  (const float*)d_in[13], (const float*)d_in[17] };
    const float* attD[3]  = { (const float*)d_in[10], (const float*)d_in[14], (const float*)d_in[18] };
    const float* bias[3]  = { (const float*)d_in[11], (const float*)d_in[15], (const float*)d_in[19] };
    const float* g1w = (const float*)d_in[20]; const float* g1b = (const float*)d_in[21];
    const float* g2w = (const float*)d_in[22]; const float* g2b = (const float*)d_in[23];
    const float* c1w = (const float*)d_in[24]; const float* c1b = (const float*)d_in[25];
    const float* c2w = (const float*)d_in[26]; const float* c2b = (const float*)d_in[27];
    const float* c3w = (const float*)d_in[28]; const float* c3b = (const float*)d_in[29];
    float* out = (float*)d_out;

    // workspace layout
    float* wsf     = (float*)d_ws;
    float* colsum  = wsf;                       // 256
    float* colsq   = wsf + 256;                 // 256
    float* scale   = wsf + 512;                 // 256
    float* shift   = wsf + 768;                 // 256
    size_t off = 1024;
    float* h_all   = wsf + off; off += (size_t)NN * HALL;     // 19.2M
    float* a_src   = wsf + off; off += (size_t)3 * NN * 2;    // 600K
    float* a_dst   = wsf + off; off += (size_t)3 * NN * 2;
    float* Wt[3];
    Wt[0] = wsf + off; off += FF * HC;
    Wt[1] = wsf + off; off += FF * HC;
    Wt[2] = wsf + off; off += FF * HC;
    float* gfeat   = wsf + off; off += (size_t)3 * GG * HC;   // 98K
    // int-typed CSR region
    int* wsi = (int*)(wsf + off);
    size_t ioff = 0;
    int* count   = wsi + ioff; ioff += (size_t)3 * NN;        // per-view in-degree
    int* offs    = wsi + ioff; ioff += (size_t)3 * (NN + 1);
    int* cursor  = wsi + ioff; ioff += (size_t)3 * NN;
    int* csr_src = wsi + ioff; ioff += (size_t)3 * ITEMS;     // 5.1M

    const float NEG_INF = -INFINITY;
    const int eb = (ITEMS + 255) / 256;

    // init scratch (0.0f bit pattern == int 0, reused for count)
    fill_kernel<<<4, 128, 0, stream>>>(colsum, 512, 0.0f);
    fill_kernel<<<96, 256, 0, stream>>>(gfeat, (long long)3 * GG * HC, NEG_INF);
    fill_kernel<<<512, 256, 0, stream>>>((float*)count, (long long)3 * NN, 0.0f);

    // BatchNorm (training-mode batch stats)
    bn_stats_kernel<<<(NN + 127) / 128, 256, 0, stream>>>(x, colsum, colsq);
    bn_finalize_kernel<<<1, 256, 0, stream>>>(colsum, colsq, bn_w, bn_b, scale, shift);

    // transpose weights for contiguous B-fragment loads
    for (int v = 0; v < 3; ++v)
        wtrans_kernel<<<(FF * HC + 255) / 256, 256, 0, stream>>>(W[v], Wt[v]);

    // fused BN + 3-view GEMM (WMMA fp32): h_all[N,192]
    gemm_bn_wmma_kernel<<<NN / 16, 128, 0, stream>>>(x, scale, shift, Wt[0], Wt[1], Wt[2], h_all);

    // attention projections
    attn_kernel<<<(3 * NN * 2 + 255) / 256, 256, 0, stream>>>(
        h_all, attS[0], attD[0], attS[1], attD[1], attS[2], attD[2], a_src, a_dst);

    // per-view: CSR build + fused gather/softmax/aggregate/ELU/pool
    for (int v = 0; v < 3; ++v) {
        int* cnt = count   + (size_t)v * NN;
        int* ofs = offs    + (size_t)v * (NN + 1);
        int* cur = cursor  + (size_t)v * NN;
        int* csr = csr_src + (size_t)v * ITEMS;
        const float* aS = a_src + (size_t)v * NN * 2;
        const float* aD = a_dst + (size_t)v * NN * 2;
        count_kernel  <<<eb, 256, 0, stream>>>(ei[v], cnt);
        scan_kernel   <<<1, 1024, 0, stream>>>(cnt, ofs, cur);
        scatter_kernel<<<eb, 256, 0, stream>>>(ei[v], cur, csr);
        gather_kernel <<<(NN * 32 + 255) / 256, 256, 0, stream>>>(
            ofs, csr, aS, aD, h_all, v * 64, bias[v], batch,
            gfeat + (size_t)v * GG * HC);
    }

    // gating + classifier
    head_kernel<<<(GG + 255) / 256, 256, 0, stream>>>(
        gfeat, g1w, g1b, g2w, g2b, c1w, c1b, c2w, c2b, c3w, c3b, out);
}